// get_pointnet2_for_instance_segmentation_model_67095979098152
// MI455X (gfx1250) — compile-verified
//
#include <hip/hip_runtime.h>
#include <cmath>

// ---------------- model constants (from reference) ----------------
constexpr int B_  = 8;
constexpr int N0  = 4096;
constexpr int S1  = 1024, NS1 = 32;
constexpr int S2  = 256,  NS2 = 64;
constexpr float EPSBN = 1e-5f;

struct BnLayer { const float *w, *b, *g, *be, *m, *v; };

typedef __attribute__((ext_vector_type(2))) float v2f;
typedef __attribute__((ext_vector_type(8))) float v8f;

// ---------------- split (B,5,N) -> xyz (B,N,3) + points (B,N,2) ----------------
__global__ void k_split(const float* __restrict__ xyz,
                        float* __restrict__ oxyz, float* __restrict__ opts) {
  int t = blockIdx.x * blockDim.x + threadIdx.x;
  if (t >= B_ * N0) return;
  int b = t / N0, n = t - b * N0;
  const float* src = xyz + (size_t)b * 5 * N0;
  float* ox = oxyz + (size_t)t * 3;
  float* op = opts + (size_t)t * 2;
  ox[0] = src[0 * N0 + n]; ox[1] = src[1 * N0 + n]; ox[2] = src[2 * N0 + n];
  op[0] = src[3 * N0 + n]; op[1] = src[4 * N0 + n];
}

// ---------------- farthest point sampling (serial chain, LDS-staged) ----------------
template <int N, int NP>
__global__ void __launch_bounds__(256) k_fps(const float* __restrict__ xyz,
                                             int* __restrict__ out) {
  constexpr int T = 256;
  constexpr int P = N / T;
  __shared__ float sx[N * 3];
  __shared__ float sval[T];
  __shared__ int   sidx[T];
  __shared__ int   sfar;
  const int b = blockIdx.x, t = threadIdx.x;
  const float* base = xyz + (size_t)b * N * 3;
  for (int i = t; i < N * 3; i += T) sx[i] = base[i];
  float dist[P];
  #pragma unroll
  for (int p = 0; p < P; ++p) dist[p] = 1e10f;
  if (t == 0) sfar = 0;
  __syncthreads();
  for (int it = 0; it < NP; ++it) {
    int far = sfar;
    if (t == 0) out[b * NP + it] = far;   // scan emits incoming 'far'
    float cx = sx[far*3+0], cy = sx[far*3+1], cz = sx[far*3+2];
    float best = -1.0f; int bi = t * P;
    #pragma unroll
    for (int p = 0; p < P; ++p) {
      int j = t * P + p;
      float dx = sx[j*3+0]-cx, dy = sx[j*3+1]-cy, dz = sx[j*3+2]-cz;
      float nd = fminf(dist[p], dx*dx + dy*dy + dz*dz);
      dist[p] = nd;
      if (nd > best) { best = nd; bi = j; }   // strict > keeps first occurrence
    }
    sval[t] = best; sidx[t] = bi;
    __syncthreads();
    for (int s = T / 2; s > 0; s >>= 1) {
      if (t < s) {
        float ov = sval[t+s]; int oi = sidx[t+s];
        if (ov > sval[t] || (ov == sval[t] && oi < sidx[t])) { sval[t] = ov; sidx[t] = oi; }
      }
      __syncthreads();
    }
    if (t == 0) sfar = sidx[0];
    __syncthreads();
  }
}

// ---------------- gather xyz rows by index ----------------
__global__ void k_gather3(const float* __restrict__ xyz, const int* __restrict__ idx,
                          float* __restrict__ out, int S, int N) {
  int t = blockIdx.x * blockDim.x + threadIdx.x;
  if (t >= B_ * S) return;
  int b = t / S;
  int j = idx[t];
  const float* src = xyz + ((size_t)b * N + j) * 3;
  float* dst = out + (size_t)t * 3;
  dst[0] = src[0]; dst[1] = src[1]; dst[2] = src[2];
}

// ---------------- ball query: first NS indices (ascending) within radius ----------------
template <int NS>
__global__ void k_ball(const float* __restrict__ xyz, const float* __restrict__ nxyz,
                       int* __restrict__ out, int S, int N, float r2) {
  int t = blockIdx.x * blockDim.x + threadIdx.x;
  if (t >= B_ * S) return;
  int b = t / S;
  const float* c = nxyz + (size_t)t * 3;
  float cx = c[0], cy = c[1], cz = c[2];
  const float* base = xyz + (size_t)b * N * 3;
  int* o = out + (size_t)t * NS;
  int cnt = 0;
  for (int j = 0; j < N && cnt < NS; ++j) {
    float dx = base[j*3+0]-cx, dy = base[j*3+1]-cy, dz = base[j*3+2]-cz;
    if (dx*dx + dy*dy + dz*dz <= r2) o[cnt++] = j;
  }
  int fill = (cnt > 0) ? o[0] : 0;
  for (; cnt < NS; ++cnt) o[cnt] = fill;
}

// ---------------- set-abstraction group: gather + 3x(GEMM+BN+ReLU) + maxpool ----------------
template <int NS, int CIN, int C1, int C2, int C3, int CPTS>
__global__ void __launch_bounds__(256) k_sa(const float* __restrict__ xyz,
                                            const float* __restrict__ pts,
                                            const float* __restrict__ nxyz,
                                            const int* __restrict__ idx,
                                            float* __restrict__ out, int S, int N,
                                            BnLayer L1, BnLayer L2, BnLayer L3) {
  constexpr int CA = (CIN > C2 ? CIN : C2);
  __shared__ float fA[NS * CA];
  __shared__ float fB[NS * C1];
  __shared__ unsigned omax[C3];
  const int g = blockIdx.x;
  const int b = g / S;
  const int t = threadIdx.x, T = blockDim.x;
  const int* gi = idx + (size_t)g * NS;
  const float* nc = nxyz + (size_t)g * 3;
  float ctr0 = nc[0], ctr1 = nc[1], ctr2 = nc[2];
  for (int e = t; e < NS * CIN; e += T) {
    int smp = e / CIN, c = e - smp * CIN;
    int j = gi[smp];
    float val;
    if (c < 3) {
      float q = xyz[((size_t)b * N + j) * 3 + c];
      val = q - (c == 0 ? ctr0 : (c == 1 ? ctr1 : ctr2));
    } else {
      val = pts[((size_t)b * N + j) * CPTS + (c - 3)];
    }
    fA[smp * CIN + c] = val;
  }
  for (int e = t; e < C3; e += T) omax[e] = 0u;
  __syncthreads();
  // layer1: fA (NSxCIN) -> fB (NSxC1)
  for (int e = t; e < NS * C1; e += T) {
    int smp = e / C1, o = e - smp * C1;
    float acc = L1.b[o];
    const float* wr = L1.w + o * CIN;
    for (int k = 0; k < CIN; ++k) acc += fA[smp * CIN + k] * wr[k];
    float sc = L1.g[o] * rsqrtf(L1.v[o] + EPSBN);
    fB[smp * C1 + o] = fmaxf((acc - L1.m[o]) * sc + L1.be[o], 0.f);
  }
  __syncthreads();
  // layer2: fB (NSxC1) -> fA (NSxC2)
  for (int e = t; e < NS * C2; e += T) {
    int smp = e / C2, o = e - smp * C2;
    float acc = L2.b[o];
    const float* wr = L2.w + o * C1;
    for (int k = 0; k < C1; ++k) acc += fB[smp * C1 + k] * wr[k];
    float sc = L2.g[o] * rsqrtf(L2.v[o] + EPSBN);
    fA[smp * C2 + o] = fmaxf((acc - L2.m[o]) * sc + L2.be[o], 0.f);
  }
  __syncthreads();
  // layer3 + fused max over samples (relu output >= 0 -> uint order == float order)
  for (int e = t; e < NS * C3; e += T) {
    int smp = e / C3, o = e - smp * C3;
    float acc = L3.b[o];
    const float* wr = L3.w + o * C2;
    for (int k = 0; k < C2; ++k) acc += fA[smp * C2 + k] * wr[k];
    float sc = L3.g[o] * rsqrtf(L3.v[o] + EPSBN);
    float val = fmaxf((acc - L3.m[o]) * sc + L3.be[o], 0.f);
    atomicMax(&omax[o], __float_as_uint(val));
  }
  __syncthreads();
  for (int e = t; e < C3; e += T)
    out[(size_t)g * C3 + e] = __uint_as_float(omax[e]);
}

// ---------------- pointwise MLP: y[p,o] = act(bn(x[p,:]·w[o,:] + b[o])) ----------------
template <int CIN, int COUT, int MODE>   // MODE 0: bias only; 1: +BN+ReLU
__global__ void k_mlp(const float* __restrict__ x, float* __restrict__ y,
                      int NP, BnLayer L) {
  int t = blockIdx.x * blockDim.x + threadIdx.x;
  if (t >= NP * COUT) return;
  int p = t / COUT, o = t - p * COUT;
  const float* xr = x + (size_t)p * CIN;
  const float* wr = L.w + o * CIN;
  float acc = L.b[o];
  for (int k = 0; k < CIN; ++k) acc += xr[k] * wr[k];
  if constexpr (MODE == 1) {
    float sc = L.g[o] * rsqrtf(L.v[o] + EPSBN);
    acc = fmaxf((acc - L.m[o]) * sc + L.be[o], 0.f);
  }
  y[(size_t)p * COUT + o] = acc;
}

// ---------------- copy points1 into feature buffer prefix ----------------
__global__ void k_copyfeat(const float* __restrict__ pts1, float* __restrict__ feat,
                           int N1, int C1, int CTOT) {
  int t = blockIdx.x * blockDim.x + threadIdx.x;
  if (t >= B_ * N1 * C1) return;
  int p = t / C1, c = t - p * C1;
  feat[(size_t)p * CTOT + c] = pts1[(size_t)p * C1 + c];
}

// ---------------- 3-NN inverse-distance interpolation into feature suffix ----------------
__global__ void k_interp(const float* __restrict__ xyz1, const float* __restrict__ xyz2,
                         const float* __restrict__ pts2, float* __restrict__ feat,
                         int N1, int N2, int C2, int CTOT, int OFF) {
  int t = blockIdx.x * blockDim.x + threadIdx.x;
  if (t >= B_ * N1) return;
  int b = t / N1;
  const float* p = xyz1 + (size_t)t * 3;
  float px = p[0], py = p[1], pz = p[2];
  const float* base = xyz2 + (size_t)b * N2 * 3;
  float d0 = 3.0e38f, d1 = 3.0e38f, d2 = 3.0e38f;
  int i0 = 0, i1 = 0, i2 = 0;
  for (int j = 0; j < N2; ++j) {
    float dx = base[j*3+0]-px, dy = base[j*3+1]-py, dz = base[j*3+2]-pz;
    float d = dx*dx + dy*dy + dz*dz;
    if (d < d0)      { d2 = d1; i2 = i1; d1 = d0; i1 = i0; d0 = d; i0 = j; }
    else if (d < d1) { d2 = d1; i2 = i1; d1 = d;  i1 = j; }
    else if (d < d2) { d2 = d;  i2 = j; }
  }
  float w0 = 1.f / (d0 + 1e-8f), w1 = 1.f / (d1 + 1e-8f), w2 = 1.f / (d2 + 1e-8f);
  float ws = 1.f / (w0 + w1 + w2);
  w0 *= ws; w1 *= ws; w2 *= ws;
  const float* r0 = pts2 + ((size_t)b * N2 + i0) * C2;
  const float* r1 = pts2 + ((size_t)b * N2 + i1) * C2;
  const float* r2 = pts2 + ((size_t)b * N2 + i2) * C2;
  float* fo = feat + (size_t)t * CTOT + OFF;
  for (int c = 0; c < C2; ++c) fo[c] = w0 * r0[c] + w1 * r1[c] + w2 * r2[c];
}

// ---------------- row norms of F_sim ----------------
__global__ void k_norm(const float* __restrict__ F, float* __restrict__ nrm) {
  int t = blockIdx.x * blockDim.x + threadIdx.x;
  if (t >= B_ * N0) return;
  const float* fr = F + (size_t)t * 16;
  float s = 0.f;
  #pragma unroll
  for (int k = 0; k < 16; ++k) s += fr[k] * fr[k];
  nrm[t] = s;
}

// ---------------- sim_mat tile via V_WMMA_F32_16X16X4_F32 (K=16 = 4 chained WMMAs) ----
// one wave per 16x16 output tile; D = F_i · F_jᵀ; epilogue: max(0, ni - 2D + nj)
__global__ void __launch_bounds__(32) k_sim(const float* __restrict__ F,
                                            const float* __restrict__ nrm,
                                            float* __restrict__ out) {
  const int jt = blockIdx.x, it = blockIdx.y, b = blockIdx.z;
  const int lane = threadIdx.x;
  const int ln = lane & 15;       // A row / B col / D col for this lane
  const int hi = lane >> 4;       // K-pair selector and D row offset (+8)
  const float* Fb = F + (size_t)b * N0 * 16;
  const float4* Ar = (const float4*)(Fb + (size_t)(it * 16 + ln) * 16);
  const float4* Br = (const float4*)(Fb + (size_t)(jt * 16 + ln) * 16);
  v8f c = {0.f, 0.f, 0.f, 0.f, 0.f, 0.f, 0.f, 0.f};
  #pragma unroll
  for (int kk = 0; kk < 4; ++kk) {      // K blocks of 4
    float4 fa = Ar[kk];
    float4 fb = Br[kk];
    v2f a, bb;
    a.x  = hi ? fa.z : fa.x;  a.y  = hi ? fa.w : fa.y;   // A: lanes0-15 K{0,1}, lanes16-31 K{2,3}
    bb.x = hi ? fb.z : fb.x;  bb.y = hi ? fb.w : fb.y;   // B mirrors (B[k][n] = F_j[n][k])
    c = __builtin_amdgcn_wmma_f32_16x16x4_f32(false, a, false, bb, (short)0, c,
                                              false, false);
  }
  const float* nb = nrm + (size_t)b * N0;
  float nj = nb[jt * 16 + ln];
  float* outb = out + (size_t)b * N0 * N0;
  #pragma unroll
  for (int v = 0; v < 8; ++v) {
    int row = v + 8 * hi;                       // C/D layout: m = v + 8*(lane>=16)
    float ni = nb[it * 16 + row];
    float val = ni - 2.f * c[v] + nj;
    outb[(size_t)(it * 16 + row) * N0 + jt * 16 + ln] = fmaxf(val, 0.f);
  }
}

// ---------------- head: cf (sigmoid) + sem (log_softmax over 13) ----------------
__global__ void k_head(const float* __restrict__ feat,
                       BnLayer cfL, const float* __restrict__ c1w, const float* __restrict__ c1b,
                       BnLayer semL, const float* __restrict__ c2w, const float* __restrict__ c2b,
                       float* __restrict__ cf_out, float* __restrict__ sem_out) {
  int t = blockIdx.x * blockDim.x + threadIdx.x;
  if (t >= B_ * N0) return;
  float f[16];
  const float* fr = feat + (size_t)t * 16;
  #pragma unroll
  for (int k = 0; k < 16; ++k) f[k] = fr[k];
  // cf branch
  float h[16];
  for (int o = 0; o < 16; ++o) {
    float acc = cfL.b[o];
    for (int k = 0; k < 16; ++k) acc += f[k] * cfL.w[o * 16 + k];
    float sc = cfL.g[o] * rsqrtf(cfL.v[o] + EPSBN);
    h[o] = fmaxf((acc - cfL.m[o]) * sc + cfL.be[o], 0.f);
  }
  float z = c1b[0];
  for (int k = 0; k < 16; ++k) z += h[k] * c1w[k];
  cf_out[t] = 1.f / (1.f + expf(-z));
  // sem branch
  float s[16];
  for (int o = 0; o < 16; ++o) {
    float acc = semL.b[o];
    for (int k = 0; k < 16; ++k) acc += f[k] * semL.w[o * 16 + k];
    float sc = semL.g[o] * rsqrtf(semL.v[o] + EPSBN);
    s[o] = fmaxf((acc - semL.m[o]) * sc + semL.be[o], 0.f);
  }
  float logit[13];
  float mx = -3.0e38f;
  for (int k = 0; k < 13; ++k) {
    float acc = c2b[k];
    for (int q = 0; q < 16; ++q) acc += s[q] * c2w[k * 16 + q];
    logit[k] = acc;
    mx = fmaxf(mx, acc);
  }
  float se = 0.f;
  for (int k = 0; k < 13; ++k) se += expf(logit[k] - mx);
  float lse = mx + logf(se);
  float* so = sem_out + (size_t)t * 13;
  for (int k = 0; k < 13; ++k) so[k] = logit[k] - lse;
}

// ================================================================
static inline int cdiv(long long a, int b) { return (int)((a + b - 1) / b); }

extern "C" void kernel_launch(void* const* d_in, const int* in_sizes, int n_in,
                              void* d_out, int out_size, void* d_ws, size_t ws_size,
                              hipStream_t stream) {
  (void)in_sizes; (void)n_in; (void)out_size; (void)ws_size;
  auto F = [&](int i) { return (const float*)d_in[i]; };
  auto mkL = [&](int base) {
    BnLayer L{F(base), F(base + 1), F(base + 2), F(base + 3), F(base + 4), F(base + 5)};
    return L;
  };
  // input tree (insertion order): xyz, dataset_D, sa1[3]x6, sa2[3]x6, fp2[2]x6, fp1[3]x6, head[18]
  const float* xyz = F(0);
  BnLayer sa1a = mkL(2),  sa1b = mkL(8),  sa1c = mkL(14);
  BnLayer sa2a = mkL(20), sa2b = mkL(26), sa2c = mkL(32);
  BnLayer fp2a = mkL(38), fp2b = mkL(44);
  BnLayer fp1a = mkL(50), fp1b = mkL(56), fp1c = mkL(62);
  BnLayer simL{F(68), F(69), F(68), F(68), F(68), F(68)};          // w,b only (MODE 0)
  BnLayer cfL{F(70), F(71), F(72), F(73), F(74), F(75)};
  const float *c1w = F(76), *c1b = F(77);
  BnLayer semL{F(78), F(79), F(80), F(81), F(82), F(83)};
  const float *c2w = F(84), *c2b = F(85);

  // workspace bump allocator (256B aligned)
  char* ws = (char*)d_ws;
  size_t off = 0;
  auto alloc = [&](size_t elems) {
    void* p = ws + off;
    off = (off + elems * 4 + 255) & ~((size_t)255);
    return p;
  };
  float* l0_xyz  = (float*)alloc((size_t)B_ * N0 * 3);
  float* l0_pts  = (float*)alloc((size_t)B_ * N0 * 2);
  int*   fps1    = (int*)  alloc((size_t)B_ * S1);
  float* l1_xyz  = (float*)alloc((size_t)B_ * S1 * 3);
  int*   idx1    = (int*)  alloc((size_t)B_ * S1 * NS1);
  float* l1_pts  = (float*)alloc((size_t)B_ * S1 * 64);
  int*   fps2    = (int*)  alloc((size_t)B_ * S2);
  float* l2_xyz  = (float*)alloc((size_t)B_ * S2 * 3);
  int*   idx2    = (int*)  alloc((size_t)B_ * S2 * NS2);
  float* l2_pts  = (float*)alloc((size_t)B_ * S2 * 256);
  float* featA   = (float*)alloc((size_t)B_ * S1 * 320);
  float* tmp1    = (float*)alloc((size_t)B_ * S1 * 64);
  float* l1_new  = (float*)alloc((size_t)B_ * S1 * 32);
  float* featB   = (float*)alloc((size_t)B_ * N0 * 34);
  float* tmp2    = (float*)alloc((size_t)B_ * N0 * 32);
  float* tmp3    = (float*)alloc((size_t)B_ * N0 * 32);
  float* l0_feat = (float*)alloc((size_t)B_ * N0 * 16);
  float* Fsim    = (float*)alloc((size_t)B_ * N0 * 16);
  float* nrm     = (float*)alloc((size_t)B_ * N0);

  float* out_sim = (float*)d_out;
  float* out_cf  = out_sim + (size_t)B_ * N0 * N0;
  float* out_sem = out_cf + (size_t)B_ * N0;

  // ---- stage 0: split ----
  k_split<<<cdiv((long long)B_ * N0, 256), 256, 0, stream>>>(xyz, l0_xyz, l0_pts);

  // ---- SA1 ----
  k_fps<N0, S1><<<B_, 256, 0, stream>>>(l0_xyz, fps1);
  k_gather3<<<cdiv((long long)B_ * S1, 256), 256, 0, stream>>>(l0_xyz, fps1, l1_xyz, S1, N0);
  k_ball<NS1><<<cdiv((long long)B_ * S1, 256), 256, 0, stream>>>(l0_xyz, l1_xyz, idx1, S1, N0, 0.01f);
  k_sa<NS1, 5, 32, 32, 64, 2><<<B_ * S1, 256, 0, stream>>>(l0_xyz, l0_pts, l1_xyz, idx1,
                                                           l1_pts, S1, N0, sa1a, sa1b, sa1c);
  // ---- SA2 ----
  k_fps<S1, S2><<<B_, 256, 0, stream>>>(l1_xyz, fps2);
  k_gather3<<<cdiv((long long)B_ * S2, 256), 256, 0, stream>>>(l1_xyz, fps2, l2_xyz, S2, S1);
  k_ball<NS2><<<cdiv((long long)B_ * S2, 256), 256, 0, stream>>>(l1_xyz, l2_xyz, idx2, S2, S1, 0.04f);
  k_sa<NS2, 67, 64, 128, 256, 64><<<B_ * S2, 256, 0, stream>>>(l1_xyz, l1_pts, l2_xyz, idx2,
                                                               l2_pts, S2, S1, sa2a, sa2b, sa2c);
  // ---- FP2: l2 -> l1 ----
  k_copyfeat<<<cdiv((long long)B_ * S1 * 64, 256), 256, 0, stream>>>(l1_pts, featA, S1, 64, 320);
  k_interp<<<cdiv((long long)B_ * S1, 256), 256, 0, stream>>>(l1_xyz, l2_xyz, l2_pts, featA,
                                                              S1, S2, 256, 320, 64);
  k_mlp<320, 64, 1><<<cdiv((long long)B_ * S1 * 64, 256), 256, 0, stream>>>(featA, tmp1, B_ * S1, fp2a);
  k_mlp<64, 32, 1><<<cdiv((long long)B_ * S1 * 32, 256), 256, 0, stream>>>(tmp1, l1_new, B_ * S1, fp2b);
  // ---- FP1: l1 -> l0 ----
  k_copyfeat<<<cdiv((long long)B_ * N0 * 2, 256), 256, 0, stream>>>(l0_pts, featB, N0, 2, 34);
  k_interp<<<cdiv((long long)B_ * N0, 256), 256, 0, stream>>>(l0_xyz, l1_xyz, l1_new, featB,
                                                              N0, S1, 32, 34, 2);
  k_mlp<34, 32, 1><<<cdiv((long long)B_ * N0 * 32, 256), 256, 0, stream>>>(featB, tmp2, B_ * N0, fp1a);
  k_mlp<32, 32, 1><<<cdiv((long long)B_ * N0 * 32, 256), 256, 0, stream>>>(tmp2, tmp3, B_ * N0, fp1b);
  k_mlp<32, 16, 1><<<cdiv((long long)B_ * N0 * 16, 256), 256, 0, stream>>>(tmp3, l0_feat, B_ * N0, fp1c);
  // ---- head ----
  k_mlp<16, 16, 0><<<cdiv((long long)B_ * N0 * 16, 256), 256, 0, stream>>>(l0_feat, Fsim, B_ * N0, simL);
  k_norm<<<cdiv((long long)B_ * N0, 256), 256, 0, stream>>>(Fsim, nrm);
  {
    dim3 g(N0 / 16, N0 / 16, B_);
    k_sim<<<g, 32, 0, stream>>>(Fsim, nrm, out_sim);
  }
  k_head<<<cdiv((long long)B_ * N0, 256), 256, 0, stream>>>(l0_feat, cfL, c1w, c1b,
                                                            semL, c2w, c2b, out_cf, out_sem);
}